// QEmbedding_65635690217583
// MI455X (gfx1250) — compile-verified
//
#include <hip/hip_runtime.h>
#include <stdint.h>

// Quantized int4 embedding gather + dequant for MI455X (gfx1250).
//
// Memory-bound: ~48KB traffic per token (16KB packed codes + 16KB fp32 scales
// in, 16KB fp32 out), ~400MB total -> ~17us at 23.3 TB/s HBM. Zero meaningful
// FLOPs, so the CDNA5 feature that matters is the async global->LDS DMA path:
// double-buffered global_load_async_to_lds_b128 (ASYNCcnt) pipelines the next
// token's 32KB row fetch behind the current token's dequant + 128-bit NT
// stores. Each wave reads back only LDS bytes its own async ops wrote, so the
// only synchronization is s_wait_asynccnt (async loads complete in order per
// wave) - no workgroup barriers.

#define GROUP_SIZE 32
#define EMBED_DIM  4096
#define TPB        256
#define VEC        (EMBED_DIM / 4)        // 1024 uint4 chunks per row
#define CHUNKS     (VEC / TPB)            // 4 chunks per thread

// Native vector types for 128-bit LDS readback / NT global stores
// (__builtin_nontemporal_store requires a native vector, not HIP_vector_type).
typedef float    v4f __attribute__((ext_vector_type(4)));
typedef uint32_t v4u __attribute__((ext_vector_type(4)));

// Async DMA: global (16B) -> LDS. First operand = LDS byte address (VGPR),
// second = 64-bit global address (VGPR pair). Tracked by ASYNCcnt.
__device__ __forceinline__ void async_load_b128(uint32_t lds_addr, const void* gaddr) {
    asm volatile("global_load_async_to_lds_b128 %0, %1, off"
                 :: "v"(lds_addr), "v"((uint64_t)(uintptr_t)gaddr)
                 : "memory");
}

// Generic LDS pointer -> 32-bit LDS offset (flat LDS addressing uses addr[31:0]).
__device__ __forceinline__ uint32_t lds_off(const void* p) {
    return (uint32_t)(uintptr_t)p;
}

__global__ __launch_bounds__(TPB) void qembed_dequant_kernel(
    const int* __restrict__ ids,          // [ntok]
    const uint32_t* __restrict__ weight,  // [V/2, D] packed nibbles (one byte per int32)
    const float* __restrict__ scale,      // [V/GROUP_SIZE, D]
    float* __restrict__ out,              // [ntok, D]
    int ntok, int nblocks)
{
    // 16B alignment required by async b128 DMA and ds_load_b128 readback.
    __shared__ __align__(16) uint32_t bufW[2][EMBED_DIM];   // 2 x 16KB packed codes
    __shared__ __align__(16) float    bufS[2][EMBED_DIM];   // 2 x 16KB scales

    const int tid = threadIdx.x;

    auto prefetch = [&](int tok, int b) {
        const int id = ids[tok];                                   // uniform scalar load
        const uint32_t* wrow = weight + (size_t)(id >> 1) * EMBED_DIM;
        const float*    srow = scale  + (size_t)(id / GROUP_SIZE) * EMBED_DIM;
#pragma unroll
        for (int c = 0; c < CHUNKS; ++c) {
            const int e = (c * TPB + tid) * 4;                     // element index (x4 per b128)
            async_load_b128(lds_off(&bufW[b][e]), wrow + e);
            async_load_b128(lds_off(&bufS[b][e]), srow + e);
        }
    };

    int t = blockIdx.x;
    if (t >= ntok) return;

    prefetch(t, 0);
    int buf = 0;

    for (; t < ntok; t += nblocks) {
        const int  next     = t + nblocks;
        const bool has_next = next < ntok;

        // Issue next token's DMA into the other buffer before draining this one.
        if (has_next) prefetch(next, buf ^ 1);

        // Async loads complete in order per wave: after issuing 8 more ops,
        // waiting until <=8 outstanding guarantees the current buffer's 8 landed.
        if (has_next) asm volatile("s_wait_asynccnt 0x8" ::: "memory");
        else          asm volatile("s_wait_asynccnt 0x0" ::: "memory");

        const int      id    = ids[t];
        const uint32_t shift = (uint32_t)(id & 1) * 4u;            // low/high nibble
        v4f* __restrict__ orow = (v4f*)(out + (size_t)t * EMBED_DIM);

#pragma unroll
        for (int c = 0; c < CHUNKS; ++c) {
            const int v = c * TPB + tid;                           // uint4 index in row
            const v4u p = ((const v4u*)bufW[buf])[v];              // ds_load_b128
            const v4f s = ((const v4f*)bufS[buf])[v];              // ds_load_b128
            v4f r;
            r.x = (float)((int)((p.x >> shift) & 15u) - 8) * s.x;
            r.y = (float)((int)((p.y >> shift) & 15u) - 8) * s.y;
            r.z = (float)((int)((p.z >> shift) & 15u) - 8) * s.z;
            r.w = (float)((int)((p.w >> shift) & 15u) - 8) * s.w;
            // Output is streamed once, never re-read: non-temporal store keeps
            // it from evicting gathered weight/scale rows (which have L2 reuse
            // on repeated token ids; scale table is 66MB vs 192MB L2).
            __builtin_nontemporal_store(r, &orow[v]);              // global_store_b128 NT
        }
        buf ^= 1;
    }
}

extern "C" void kernel_launch(void* const* d_in, const int* in_sizes, int n_in,
                              void* d_out, int out_size, void* d_ws, size_t ws_size,
                              hipStream_t stream) {
    const int*      ids    = (const int*)d_in[0];        // [4,2048] int32
    const uint32_t* weight = (const uint32_t*)d_in[1];   // [V/2, 4096] int32 (packed bytes)
    const float*    scale  = (const float*)d_in[2];      // [V/32, 4096] fp32
    float*          out    = (float*)d_out;              // [4,2048,4096] fp32

    const int ntok   = in_sizes[0];                      // 8192 tokens
    const int blocks = ntok < 1024 ? ntok : 1024;        // 8 tokens per block

    qembed_dequant_kernel<<<blocks, TPB, 0, stream>>>(ids, weight, scale, out, ntok, blocks);
}